// HbvMoeV2_37529424233288
// MI455X (gfx1250) — compile-verified
//
#include <hip/hip_runtime.h>
#include <math.h>

// HBV-MoE fused kernel for gfx1250 (MI455X, wave32).
// One thread per (basin b, component m) channel: 1024*16 = 16384 threads (512 wave32s).
// Latency-bound (2000-step serial chain, <1 wave/SIMD), so:
//  - x_phy staged tile-wise into LDS, DOUBLE-BUFFERED: tile i+1 prefetched with
//    GLOBAL_LOAD_ASYNC_TO_LDS_B128 (ASYNCcnt) while tile i computes; drained with
//    s_wait_asynccnt before the end-of-tile barrier. Fallback: register-pipelined b128 loads.
//  - 2000-step scan fully in registers (5 states + 15-deep FIR ring + params + weights).
//  - Routing FIR (15 taps, per-channel gamma weights) fused per lane.
//  - Mean over the 16 NMUL components via V_WMMA_F32_16X16X4_F32 every 16 steps:
//    A = 16(time) x 16(m) Qrout tile (4 K=4 chunks), B = constant 1/16 matrix
//    (layout-independent), D column 0 = flow[t0..t0+15, b].

#define T_STEPS 2000
#define BGRID   1024
#define NMUL    16
#define N_PHY   16
#define PSTRIDE (N_PHY * NMUL + 2 * NMUL)   // 288
#define LENF    15
#define TILE    16
#define BPB     16                          // basins per block (256 threads)
#define NEARZERO 1e-5f
#define XB_FLOATS (TILE * BPB * 3)          // 768 floats per buffer
#define XB_VEC4   (XB_FLOATS / 4)           // 192 b128 transfers per buffer

typedef __attribute__((ext_vector_type(2))) float v2f;
typedef __attribute__((ext_vector_type(8))) float v8f;

#if __has_builtin(__builtin_amdgcn_global_load_async_to_lds_b128) && \
    __has_builtin(__builtin_amdgcn_s_wait_asynccnt)
#define USE_ASYNC_LDS 1
typedef int v4i __attribute__((vector_size(16)));               // matches builtin param type
typedef __attribute__((address_space(1))) v4i* as1_v4i;         // global int4*
typedef __attribute__((address_space(3))) v4i* as3_v4i;         // LDS int4* (32-bit ptr)
#else
#define USE_ASYNC_LDS 0
#endif

__launch_bounds__(256, 1)
__global__ void hbv_fused_wmma_kernel(const float* __restrict__ x_phy,
                                      const float* __restrict__ ps,
                                      float* __restrict__ out)
{
    __shared__ float xbuf[2][XB_FLOATS];              // double-buffered input tiles, 6 KB
    __shared__ float ldsQ[BPB][TILE][NMUL + 1];       // WMMA transpose buffer, ~17 KB

    const int tid     = threadIdx.x;
    const int m       = tid & 15;          // NMUL component
    const int b_local = tid >> 4;          // basin within block (0..15)
    const int lane    = tid & 31;
    const int waveId  = tid >> 5;          // wave covers b_local {2w, 2w+1}
    const int half    = lane >> 4;         // wmma half-wave selector
    const int trow    = lane & 15;         // wmma A-matrix row (time)
    const int b0      = blockIdx.x * BPB;
    const int b       = b0 + b_local;

    // staging role for this thread (192 of 256 threads move one float4 per tile)
    const bool stager = (tid < XB_VEC4);
    const int  s_tr   = tid / 12;          // time row in tile
    const int  s_q    = tid % 12;          // float4 within 48-float row

    // ---------------- parameter load + bound scaling ----------------
    const float lo[N_PHY] = {1.0f, 50.0f,  0.05f,0.01f,0.001f,0.2f, 0.0f, 0.0f, -2.5f,0.5f, 0.0f,0.0f,0.3f,0.0f,0.0f, 0.0f};
    const float hi[N_PHY] = {6.0f,1000.0f, 0.9f, 0.5f, 0.2f,  1.0f,10.0f,100.0f, 2.5f,10.0f,0.1f,0.2f,5.0f,1.0f,20.0f,2500.0f};
    float p[N_PHY];
    #pragma unroll
    for (int i = 0; i < N_PHY; ++i) {
        float raw = ps[(size_t)b * PSTRIDE + i * NMUL + m];
        p[i] = fmaf(raw, hi[i] - lo[i], lo[i]);
    }
    const float pBETA = p[0], pFC = p[1], pK0 = p[2], pK1 = p[3], pK2 = p[4];
    const float pLP = p[5], pPERC = p[6], pUZL = p[7], pTT = p[8], pCFMAX = p[9];
    const float pCFRCFMAX = p[10] * p[9], pCWH = p[11], pBETAET = p[12], pC = p[13];
    const float invFC = 1.0f / pFC;
    const float invLPFC = 1.0f / (pLP * pFC);

    // ---------------- gamma routing weights (normalized; 1/16 mean folded into WMMA B) ----
    float rout_a = ps[(size_t)b * PSTRIDE + N_PHY * NMUL + m] * 2.9f;
    float rout_b = ps[(size_t)b * PSTRIDE + N_PHY * NMUL + NMUL + m] * 6.5f;
    float aa    = fmaxf(rout_a, 0.0f) + 0.1f;
    float theta = fmaxf(rout_b, 0.0f) + 0.5f;
    float wk[LENF];
    {
        float lg = lgammaf(aa);
        float lth = logf(theta);
        float invth = 1.0f / theta;
        float wsum = 0.0f;
        #pragma unroll
        for (int k = 0; k < LENF; ++k) {
            float tk = (float)k + 0.5f;
            float lw = fmaf(aa - 1.0f, logf(tk), -lg - aa * lth - tk * invth);
            wk[k] = expf(lw);
            wsum += wk[k];
        }
        float winv = 1.0f / wsum;
        #pragma unroll
        for (int k = 0; k < LENF; ++k) wk[k] *= winv;
    }

    // ---------------- state (registers) ----------------
    float SNOWPACK = NEARZERO, MELTWATER = NEARZERO, SM = NEARZERO, SUZ = NEARZERO, SLZ = NEARZERO;
    float hist[LENF];
    #pragma unroll
    for (int k = 0; k < LENF; ++k) hist[k] = 0.0f;   // zero-padded history (matches jnp.pad)

    v2f Bc; Bc.x = 0.0625f; Bc.y = 0.0625f;          // constant 1/16 B matrix (layout-proof)

    // ---------------- prologue: stage tile 0 into buffer 0 ----------------
    if (stager) {
        const float* src = x_phy + ((size_t)s_tr * BGRID + b0) * 3 + s_q * 4;
#if USE_ASYNC_LDS
        __builtin_amdgcn_global_load_async_to_lds_b128(
            (as1_v4i)(uintptr_t)src,
            (as3_v4i)(uint32_t)(uintptr_t)&xbuf[0][s_tr * 48 + s_q * 4],
            0, 0);
#else
        *(float4*)(&xbuf[0][s_tr * 48 + s_q * 4]) = *(const float4*)src;
#endif
    }
#if USE_ASYNC_LDS
    __builtin_amdgcn_s_wait_asynccnt(0);
#endif
    __syncthreads();

    // ---------------- time loop: 125 tiles of 16 steps, software-pipelined ----------------
    int cur = 0;
    for (int t0 = 0; t0 < T_STEPS; t0 += TILE) {
        const int nxt = cur ^ 1;
        const bool have_next = (t0 + TILE) < T_STEPS;

#if !USE_ASYNC_LDS
        float4 pre;   // register-pipelined prefetch fallback
#endif
        if (have_next && stager) {
            const float* src = x_phy + ((size_t)(t0 + TILE + s_tr) * BGRID + b0) * 3 + s_q * 4;
#if USE_ASYNC_LDS
            // prefetch tile i+1 straight into the other LDS buffer (ASYNCcnt path)
            __builtin_amdgcn_global_load_async_to_lds_b128(
                (as1_v4i)(uintptr_t)src,
                (as3_v4i)(uint32_t)(uintptr_t)&xbuf[nxt][s_tr * 48 + s_q * 4],
                0, 0);
#else
            pre = *(const float4*)src;
#endif
        }

        float qr16[TILE];
        #pragma unroll
        for (int j = 0; j < TILE; ++j) {
            const float Pm   = xbuf[cur][j * 48 + b_local * 3 + 0];
            const float Tc   = xbuf[cur][j * 48 + b_local * 3 + 1];
            const float PETm = xbuf[cur][j * 48 + b_local * 3 + 2];

            // --- snow routine ---
            float rain = (Tc >= pTT) ? Pm : 0.0f;
            float snow = Pm - rain;
            SNOWPACK += snow;
            float melt = fminf(fmaxf(pCFMAX * (Tc - pTT), 0.0f), SNOWPACK);
            MELTWATER += melt;  SNOWPACK -= melt;
            float refreeze = fminf(fmaxf(pCFRCFMAX * (pTT - Tc), 0.0f), MELTWATER);
            SNOWPACK += refreeze;  MELTWATER -= refreeze;
            float tosoil = fmaxf(MELTWATER - pCWH * SNOWPACK, 0.0f);
            MELTWATER -= tosoil;

            // --- soil moisture ---
            float sw = fminf(__expf(pBETA * __logf(SM * invFC)), 1.0f);   // (SM/FC)^BETA clipped
            float inflow = rain + tosoil;
            float recharge = inflow * sw;
            SM += inflow - recharge;
            float excess = fmaxf(SM - pFC, 0.0f);
            SM -= excess;
            float capillary = fminf(SLZ, pC * SLZ * (1.0f - fminf(SM * invFC, 1.0f)));
            SM += capillary;  SLZ -= capillary;
            float ef = fminf(SM * invLPFC, 1.0f);
            ef = __expf(pBETAET * __logf(ef));
            float ETact = fminf(PETm * ef, SM);
            SM = fmaxf(SM - ETact, NEARZERO);

            // --- response routine ---
            SUZ += recharge + excess;
            float PERC = fminf(SUZ, pPERC);
            SUZ -= PERC;
            float Q0 = pK0 * fmaxf(SUZ - pUZL, 0.0f);
            SUZ -= Q0;
            float Q1 = pK1 * SUZ;
            SUZ -= Q1;
            SLZ += PERC;
            float Q2 = pK2 * SLZ;
            SLZ -= Q2;
            float Qsim = Q0 + Q1 + Q2;

            // --- fused 15-tap gamma FIR (ring buffer in registers) ---
            #pragma unroll
            for (int k = LENF - 1; k > 0; --k) hist[k] = hist[k - 1];
            hist[0] = Qsim;
            float qr = 0.0f;
            #pragma unroll
            for (int k = 0; k < LENF; ++k) qr = fmaf(wk[k], hist[k], qr);
            qr16[j] = qr;
        }

        // ---- transpose Qrout tile into LDS for WMMA A-matrix layout ----
        #pragma unroll
        for (int j = 0; j < TILE; ++j) ldsQ[b_local][j][m] = qr16[j];
        __syncthreads();

        // ---- mean over m via V_WMMA_F32_16X16X4_F32: D = Qtile(16x16) * (1/16)ones ----
        // A 16x4 f32 layout: lanes 0-15 -> K=0(v0),K=1(v1); lanes 16-31 -> K=2(v0),K=3(v1)
        v8f cc[2];
        #pragma unroll
        for (int s = 0; s < 2; ++s) {
            const int bs = waveId * 2 + s;
            v8f c = {};
            #pragma unroll
            for (int ch = 0; ch < 4; ++ch) {
                const int mc = ch * 4 + half * 2;
                v2f a;
                a.x = ldsQ[bs][trow][mc + 0];
                a.y = ldsQ[bs][trow][mc + 1];
                c = __builtin_amdgcn_wmma_f32_16x16x4_f32(
                        /*neg_a=*/false, a, /*neg_b=*/false, Bc,
                        /*c_mod=*/(short)0, c, /*reuse_a=*/false, /*reuse_b=*/false);
            }
            cc[s] = c;
        }

        // ---- D layout: VGPR r holds M=r (lanes 0-15) / M=r+8 (lanes 16-31), N=lane%16.
        // Column 0 lives in lanes 0 and 16 -> those lanes store flow[t0+M, b].
        if ((lane & 15) == 0) {
            const int rbase = half * 8;
            const int bA = b0 + waveId * 2;
            #pragma unroll
            for (int r = 0; r < 8; ++r) {
                out[(size_t)(t0 + rbase + r) * BGRID + bA]     = cc[0][r];
                out[(size_t)(t0 + rbase + r) * BGRID + bA + 1] = cc[1][r];
            }
        }

#if !USE_ASYNC_LDS
        if (have_next && stager) {
            *(float4*)(&xbuf[nxt][s_tr * 48 + s_q * 4]) = pre;   // commit prefetch late
        }
#else
        __builtin_amdgcn_s_wait_asynccnt(0);   // tile i+1 resident before anyone reads it
#endif
        __syncthreads();   // xbuf[nxt] ready / ldsQ reuse fence
        cur = nxt;
    }
}

extern "C" void kernel_launch(void* const* d_in, const int* in_sizes, int n_in,
                              void* d_out, int out_size, void* d_ws, size_t ws_size,
                              hipStream_t stream) {
    (void)in_sizes; (void)n_in; (void)d_ws; (void)ws_size; (void)out_size;
    const float* x_phy = (const float*)d_in[0];   // (2000, 1024, 3) f32
    const float* ps    = (const float*)d_in[1];   // (1024, 288)     f32
    float* out         = (float*)d_out;           // (2000, 1024)    f32

    dim3 grid(BGRID / BPB);   // 64 blocks
    dim3 block(256);          // 8 wave32s, 16 basins/block
    hbv_fused_wmma_kernel<<<grid, block, 0, stream>>>(x_phy, ps, out);
}